// PtrNetModel_87488483820151
// MI455X (gfx1250) — compile-verified
//
#include <hip/hip_runtime.h>
#include <hip/hip_bf16.h>
#include <math.h>

// ---- problem constants (match reference) ----
#define N_NODES 2049      // 1 + T*A
#define T_TOK   256
#define ARCS    8
#define H_ENC   256
#define DH_DEC  512
#define D_IN    800       // 300+300+100+100
#define G_ENC   1024      // 4*H
#define G_DEC   2048      // 4*DH

typedef __attribute__((ext_vector_type(2))) float v2f;
typedef __attribute__((ext_vector_type(8))) float v8f;

__device__ __forceinline__ float sigf(float x) { return 1.0f / (1.0f + expf(-x)); }

// ---------------------------------------------------------------------------
// 1) Embedding gather: emb[n][d], d<300 form, <600 lemma, <700 tag, <800 mean(feats)
// ---------------------------------------------------------------------------
__global__ void k_gather(const int* __restrict__ lat,
                         const float* __restrict__ form,
                         const float* __restrict__ lemma,
                         const float* __restrict__ tag,
                         const float* __restrict__ feats,
                         float* __restrict__ emb) {
    int idx = blockIdx.x * 256 + threadIdx.x;
    if (idx >= N_NODES * D_IN) return;
    int n = idx / D_IN, d = idx - n * D_IN;
    const int* row = lat + n * 11;
    float v;
    if (d < 300) {
        v = form[(size_t)row[0] * 300 + d];
    } else if (d < 600) {
        v = lemma[(size_t)row[1] * 300 + (d - 300)];
    } else if (d < 700) {
        v = tag[row[2] * 100 + (d - 600)];
    } else {
        float s = 0.0f;
        #pragma unroll
        for (int j = 0; j < 6; ++j) s += feats[row[3 + j] * 100 + (d - 700)];
        v = s * (1.0f / 6.0f);
    }
    emb[idx] = v;
}

// ---------------------------------------------------------------------------
// 2) Generic transpose: in[R][K] -> out[K][R]
// ---------------------------------------------------------------------------
__global__ void k_transpose(const float* __restrict__ in, float* __restrict__ out,
                            int R, int K) {
    int idx = blockIdx.x * 256 + threadIdx.x;
    if (idx >= R * K) return;
    int r = idx / K, k = idx - r * K;
    out[(size_t)k * R + r] = in[idx];
}

// ---------------------------------------------------------------------------
// 3) WMMA fp32 GEMM: C[M][Nc] = A[M][K] * Bt[K][Nc] + bias[Nc]
//    One wave per 16x64 strip (4 accumulators) so the A fragment is reused 4x.
//    K stepped by 4 via V_WMMA_F32_16X16X4_F32.
//    Fragment layouts per CDNA5 ISA 7.12.2:
//      A 16x4 : lane L holds row (L&15); VGPR0/1 = K {0,1} (lanes 0-15) or {2,3} (16-31)
//      B 4x16 : VGPR r, lanes 0-15 = K row r, lanes 16-31 = K row r+2; col = lane&15
//      C 16x16: VGPR r, lanes 0-15 = M row r, lanes 16-31 = M row r+8; col = lane&15
// ---------------------------------------------------------------------------
__global__ void k_gemm_wmma(const float* __restrict__ A,
                            const float* __restrict__ Bt,
                            const float* __restrict__ bias,
                            float* __restrict__ C,
                            int M, int K, int Nc) {
    int stripsN = Nc >> 6;                       // Nc is a multiple of 64
    int tileN = (blockIdx.x % stripsN) << 6;
    int tileM = (blockIdx.x / stripsN) << 4;
    int lane = threadIdx.x;                      // wave32
    int half = lane >> 4;                        // 0 = lanes 0-15, 1 = lanes 16-31
    int l15  = lane & 15;

    int rowA = tileM + l15;
    if (rowA >= M) rowA = M - 1;                 // clamp loads; stores guarded below
    int col0 = tileN + l15;

    v8f acc0 = {}, acc1 = {}, acc2 = {}, acc3 = {};
    for (int k = 0; k < K; k += 4) {
        const float* ar = A + (size_t)rowA * K + (k + half * 2);
        v2f a; a.x = ar[0]; a.y = ar[1];         // contiguous -> global_load_b64
        const float* br = Bt + (size_t)(k + half * 2) * Nc + col0;
        v2f b0, b1, b2, b3;
        b0.x = br[0];  b0.y = br[Nc];
        b1.x = br[16]; b1.y = br[Nc + 16];
        b2.x = br[32]; b2.y = br[Nc + 32];
        b3.x = br[48]; b3.y = br[Nc + 48];
        acc0 = __builtin_amdgcn_wmma_f32_16x16x4_f32(false, a, false, b0, (short)0, acc0, false, false);
        acc1 = __builtin_amdgcn_wmma_f32_16x16x4_f32(false, a, false, b1, (short)0, acc1, false, false);
        acc2 = __builtin_amdgcn_wmma_f32_16x16x4_f32(false, a, false, b2, (short)0, acc2, false, false);
        acc3 = __builtin_amdgcn_wmma_f32_16x16x4_f32(false, a, false, b3, (short)0, acc3, false, false);
    }

    int rowBase = tileM + (half ? 8 : 0);
    #pragma unroll
    for (int tn = 0; tn < 4; ++tn) {
        int col = col0 + tn * 16;
        float bb = bias ? bias[col] : 0.0f;
        const v8f* accp = tn == 0 ? &acc0 : tn == 1 ? &acc1 : tn == 2 ? &acc2 : &acc3;
        #pragma unroll
        for (int r = 0; r < 8; ++r) {
            int row = rowBase + r;
            if (row < M) C[(size_t)row * Nc + col] = (*accp)[r] + bb;
        }
    }
}

// ---------------------------------------------------------------------------
// 4) Persistent LSTM scan (encoder). 2 blocks: block 0 = forward, block 1 = backward.
//    X = precomputed Wih@x + b, [N][1024]. WhhT = [256][1024]. 256 threads; thread t
//    owns hidden unit t (gate rows t, 256+t, 512+t, 768+t).
// ---------------------------------------------------------------------------
__global__ void k_lstm_scan(const float* __restrict__ Xf, const float* __restrict__ WhhTf,
                            float* __restrict__ hsf,
                            const float* __restrict__ Xb, const float* __restrict__ WhhTb,
                            float* __restrict__ hsb,
                            float* __restrict__ h0, float* __restrict__ c0) {
    __shared__ float h_sh[H_ENC];
    const int t = threadIdx.x;
    const int fwd = (blockIdx.x == 0);
    const float* X    = fwd ? Xf : Xb;
    const float* WhhT = fwd ? WhhTf : WhhTb;
    float*       hs   = fwd ? hsf : hsb;
    const int hOff    = fwd ? 0 : H_ENC;

    float c = 0.0f;
    h_sh[t] = 0.0f;
    __syncthreads();

    for (int s = 0; s < N_NODES; ++s) {
        int n = fwd ? s : (N_NODES - 1 - s);
        // prefetch next step's precomputed gate row while we grind the GEMV
        int n_next = fwd ? (s + 1 < N_NODES ? s + 1 : n) : (n > 0 ? n - 1 : 0);
        __builtin_prefetch(X + (size_t)n_next * G_ENC + t, 0, 1);

        const float* xr = X + (size_t)n * G_ENC;
        float gi = xr[t];
        float gf = xr[H_ENC + t];
        float gg = xr[2 * H_ENC + t];
        float go = xr[3 * H_ENC + t];
        for (int j = 0; j < H_ENC; ++j) {
            float hj = h_sh[j];
            const float* w = WhhT + (size_t)j * G_ENC;
            gi = fmaf(w[t], hj, gi);
            gf = fmaf(w[H_ENC + t], hj, gf);
            gg = fmaf(w[2 * H_ENC + t], hj, gg);
            go = fmaf(w[3 * H_ENC + t], hj, go);
        }
        c = sigf(gf) * c + sigf(gi) * tanhf(gg);
        float h = sigf(go) * tanhf(c);
        __syncthreads();            // all reads of previous h done
        h_sh[t] = h;
        hs[(size_t)n * H_ENC + t] = h;
        __syncthreads();
    }
    h0[hOff + t] = h_sh[t];
    c0[hOff + t] = c;
}

// ---------------------------------------------------------------------------
// 5) dec_const[row] = dec_b[row] + dec_whh[row] . h0   (row < 2048)
// ---------------------------------------------------------------------------
__global__ void k_dec_const(const float* __restrict__ dec_whh,
                            const float* __restrict__ dec_b,
                            const float* __restrict__ h0,
                            float* __restrict__ dc) {
    int row = blockIdx.x * 256 + threadIdx.x;
    if (row >= G_DEC) return;
    float s = dec_b[row];
    const float* w = dec_whh + (size_t)row * DH_DEC;
    for (int k = 0; k < DH_DEC; ++k) s = fmaf(w[k], h0[k], s);
    dc[row] = s;
}

// ---------------------------------------------------------------------------
// 6) Decoder: 256 pointer steps. 512 threads; thread t owns hidden unit t of DH.
//    Candidates at step i are the 8 nodes with tok==i: [1+(i-1)*8, ... +7];
//    gold is candidate 0. Masked log_softmax == logsumexp over these 8.
//    out[0..256] = pred indices (as float), out[257] = loss.
// ---------------------------------------------------------------------------
__global__ void k_decoder(const float* __restrict__ Xd,       // [N][2048] (no bias)
                          const float* __restrict__ dc,       // [2048]
                          const float* __restrict__ c0,       // [512]
                          const float* __restrict__ hsf,      // [N][256]
                          const float* __restrict__ hsb,      // [N][256]
                          float* __restrict__ out) {
    __shared__ float h_sh[DH_DEC];
    __shared__ float sc[ARCS];
    __shared__ int   ptr_sh;
    __shared__ float loss_sh;
    const int t = threadIdx.x;
    if (t == 0) { ptr_sh = 0; loss_sh = 0.0f; }
    __syncthreads();

    for (int i = 1; i <= T_TOK; ++i) {
        int p = ptr_sh;
        const float* xr = Xd + (size_t)p * G_DEC;
        float gi = xr[t] + dc[t];
        float gf = xr[DH_DEC + t] + dc[DH_DEC + t];
        float gg = xr[2 * DH_DEC + t] + dc[2 * DH_DEC + t];
        float go = xr[3 * DH_DEC + t] + dc[3 * DH_DEC + t];
        float c = sigf(gf) * c0[t] + sigf(gi) * tanhf(gg);
        h_sh[t] = sigf(go) * tanhf(c);
        __syncthreads();

        // 8 waves -> 8 candidate scores (512-d dot each), wave32 shuffle reduce
        int w = t >> 5, lane = t & 31;
        if (w < ARCS) {
            int node = 1 + (i - 1) * ARCS + w;
            const float* ef = hsf + (size_t)node * H_ENC;
            const float* eb = hsb + (size_t)node * H_ENC;
            float s = 0.0f;
            #pragma unroll
            for (int k = 0; k < 8; ++k) {
                s = fmaf(ef[lane + 32 * k], h_sh[lane + 32 * k], s);
                s = fmaf(eb[lane + 32 * k], h_sh[H_ENC + lane + 32 * k], s);
            }
            #pragma unroll
            for (int off = 16; off > 0; off >>= 1) s += __shfl_down(s, off, 32);
            if (lane == 0) sc[w] = s;
        }
        __syncthreads();

        if (t == 0) {
            float m = sc[0]; int am = 0;
            #pragma unroll
            for (int a = 1; a < ARCS; ++a)
                if (sc[a] > m) { m = sc[a]; am = a; }   // first-max tie-break
            float sum = 0.0f;
            #pragma unroll
            for (int a = 0; a < ARCS; ++a) sum += expf(sc[a] - m);
            float lse = m + logf(sum);
            loss_sh += (lse - sc[0]);                    // -logp(gold)
            int ptr = 1 + (i - 1) * ARCS + am;
            out[i] = (float)ptr;
            ptr_sh = ptr;
        }
        __syncthreads();
    }
    if (t == 0) {
        out[0] = 0.0f;                                   // gold[0] == node 0
        out[T_TOK + 1] = loss_sh / (float)T_TOK;
    }
}

// ---------------------------------------------------------------------------
// Host-side launch
// ---------------------------------------------------------------------------
extern "C" void kernel_launch(void* const* d_in, const int* in_sizes, int n_in,
                              void* d_out, int out_size, void* d_ws, size_t ws_size,
                              hipStream_t stream) {
    (void)in_sizes; (void)n_in; (void)out_size; (void)ws_size;

    const int*   lat       = (const int*)  d_in[0];
    const float* form_emb  = (const float*)d_in[2];
    const float* lemma_emb = (const float*)d_in[3];
    const float* tag_emb   = (const float*)d_in[4];
    const float* feats_emb = (const float*)d_in[5];
    const float* wih_f     = (const float*)d_in[6];
    const float* whh_f     = (const float*)d_in[7];
    const float* b_f       = (const float*)d_in[8];
    const float* wih_b     = (const float*)d_in[9];
    const float* whh_b     = (const float*)d_in[10];
    const float* b_b       = (const float*)d_in[11];
    const float* dec_wih   = (const float*)d_in[12];
    const float* dec_whh   = (const float*)d_in[13];
    const float* dec_b     = (const float*)d_in[14];
    float* out = (float*)d_out;

    // ---- workspace carve-up (fp32, 256-element aligned) ----
    float* ws = (float*)d_ws;
    size_t off = 0;
    auto alloc = [&](size_t nfl) {
        float* p = ws + off;
        off += (nfl + 255) & ~(size_t)255;
        return p;
    };
    float* emb    = alloc((size_t)N_NODES * D_IN);
    float* wihfT  = alloc((size_t)D_IN * G_ENC);
    float* wihbT  = alloc((size_t)D_IN * G_ENC);
    float* wihdT  = alloc((size_t)D_IN * G_DEC);
    float* whhfT  = alloc((size_t)H_ENC * G_ENC);
    float* whhbT  = alloc((size_t)H_ENC * G_ENC);
    float* Xf     = alloc((size_t)N_NODES * G_ENC);
    float* Xb     = alloc((size_t)N_NODES * G_ENC);
    float* Xd     = alloc((size_t)N_NODES * G_DEC);
    float* hsf    = alloc((size_t)N_NODES * H_ENC);
    float* hsb    = alloc((size_t)N_NODES * H_ENC);
    float* h0     = alloc(DH_DEC);
    float* c0     = alloc(DH_DEC);
    float* dcnst  = alloc(G_DEC);

    // 1) gather embeddings
    {
        int tot = N_NODES * D_IN;
        k_gather<<<(tot + 255) / 256, 256, 0, stream>>>(
            lat, form_emb, lemma_emb, tag_emb, feats_emb, emb);
    }
    // 2) transpose weights for coalesced WMMA B-operand / scan reads
    {
        int tot;
        tot = G_ENC * D_IN;
        k_transpose<<<(tot + 255) / 256, 256, 0, stream>>>(wih_f, wihfT, G_ENC, D_IN);
        k_transpose<<<(tot + 255) / 256, 256, 0, stream>>>(wih_b, wihbT, G_ENC, D_IN);
        tot = G_DEC * D_IN;
        k_transpose<<<(tot + 255) / 256, 256, 0, stream>>>(dec_wih, wihdT, G_DEC, D_IN);
        tot = G_ENC * H_ENC;
        k_transpose<<<(tot + 255) / 256, 256, 0, stream>>>(whh_f, whhfT, G_ENC, H_ENC);
        k_transpose<<<(tot + 255) / 256, 256, 0, stream>>>(whh_b, whhbT, G_ENC, H_ENC);
    }
    // 3) big WMMA GEMMs: Xf, Xb (+enc bias), Xd (bias folded into dec_const)
    {
        int tilesM = (N_NODES + 15) / 16;   // 129
        k_gemm_wmma<<<tilesM * (G_ENC / 64), 32, 0, stream>>>(
            emb, wihfT, b_f, Xf, N_NODES, D_IN, G_ENC);
        k_gemm_wmma<<<tilesM * (G_ENC / 64), 32, 0, stream>>>(
            emb, wihbT, b_b, Xb, N_NODES, D_IN, G_ENC);
        k_gemm_wmma<<<tilesM * (G_DEC / 64), 32, 0, stream>>>(
            emb, wihdT, (const float*)nullptr, Xd, N_NODES, D_IN, G_DEC);
    }
    // 4) encoder scans (fwd + bwd concurrently, 2 blocks)
    k_lstm_scan<<<2, H_ENC, 0, stream>>>(Xf, whhfT, hsf, Xb, whhbT, hsb, h0, c0);
    // 5) dec_const = dec_whh @ h0 + dec_b
    k_dec_const<<<(G_DEC + 255) / 256, 256, 0, stream>>>(dec_whh, dec_b, h0, dcnst);
    // 6) decoder pointer loop
    k_decoder<<<1, DH_DEC, 0, stream>>>(Xd, dcnst, c0, hsf, hsb, out);
}